// GCN_small_49581102465677
// MI455X (gfx1250) — compile-verified
//
#include <hip/hip_runtime.h>
#include <hip/hip_bf16.h>
#include <stdint.h>

typedef float v2f __attribute__((ext_vector_type(2)));
typedef float v8f __attribute__((ext_vector_type(8)));

#define GCN_N_NODES 100000
#define GCN_N_EDGES 3200000

// ---------------- degree / normalization ----------------

__global__ void deg_init_kernel(float* __restrict__ deg, int n) {
    int i = blockIdx.x * blockDim.x + threadIdx.x;
    if (i < n) deg[i] = 1.0f;   // self-loop weight
}

__global__ void deg_scatter_kernel(const long long* __restrict__ dst64,
                                   const float* __restrict__ w,
                                   float* __restrict__ deg, int ne) {
    int e = blockIdx.x * blockDim.x + threadIdx.x;
    if (e < ne) atomicAdd(&deg[(int)dst64[e]], w[e]);
}

__global__ void dis_kernel(const float* __restrict__ deg,
                           float* __restrict__ dis, int n) {
    int i = blockIdx.x * blockDim.x + threadIdx.x;
    if (i >= n) return;
    float d = deg[i];
    dis[i] = (d > 0.0f) ? rsqrtf(d) : 0.0f;
}

// Fuse int64->int32 edge conversion with norm computation.
__global__ void edge_norm_kernel(const long long* __restrict__ ei,
                                 const float* __restrict__ w,
                                 const float* __restrict__ dis,
                                 int* __restrict__ src32,
                                 int* __restrict__ dst32,
                                 float* __restrict__ norm, int ne) {
    int e = blockIdx.x * blockDim.x + threadIdx.x;
    if (e >= ne) return;
    int s = (int)ei[e];
    int d = (int)ei[(size_t)GCN_N_EDGES + e];
    src32[e] = s;
    dst32[e] = d;
    norm[e] = dis[s] * w[e] * dis[d];
}

// ---------------- dense transform: H = X @ W via V_WMMA_F32_16X16X4_F32 ----
// Compile-time K/NC: fully unrolled wmma chain, 32-bit address math.
// W staged zero-padded in LDS -> unconditional B-fragment ds_loads.
// A-fragment tail handled with clamped index + cndmask (no exec divergence;
// EXEC stays all-1s at the WMMA as the ISA requires).
// fp32 A frag: lanes 0-15 hold K={0,1}, lanes 16-31 K={2,3} (2 VGPRs).
// C/D: 8 VGPRs; lanes 0-15 -> M=r, lanes 16-31 -> M=r+8.

template <int K, int NC>
__global__ __launch_bounds__(128) void gemm_wmma_f32_kernel(
    const float* __restrict__ X, const float* __restrict__ W,
    float* __restrict__ H, int rtiles) {
    constexpr int KITERS = (K + 3) / 4;
    constexpr int KPAD   = KITERS * 4;
    constexpr int CT     = (NC + 15) / 16;
    constexpr int NCPAD  = CT * 16;

    __shared__ float Wlds[KPAD][NCPAD];
    for (int i = threadIdx.x; i < KPAD * NCPAD; i += blockDim.x) {
        int k = i / NCPAD;
        int n = i - k * NCPAD;
        Wlds[k][n] = (k < K && n < NC) ? W[k * NC + n] : 0.0f;
    }
    __syncthreads();

    int wave = blockIdx.x * (blockDim.x >> 5) + (threadIdx.x >> 5);
    if (wave >= rtiles * CT) return;
    int lane = threadIdx.x & 31;
    int rt = wave / CT;
    int ct = wave - rt * CT;
    int lm = lane & 15;   // row-in-tile (A) / col-in-tile (B,D)
    int kh = lane >> 4;   // K-pair select: 0 -> K{0,1}, 1 -> K{2,3}

    const float* xrow = X + (rt * 16 + lm) * K;   // < 2^31, 32-bit math
    int bn = ct * 16 + lm;

    // Preload fragments (branchless tail for A; B padded with zeros in LDS).
    float a0[KITERS], a1[KITERS], b0[KITERS], b1[KITERS];
#pragma unroll
    for (int kk = 0; kk < KITERS; ++kk) {
        int k0 = kk * 4 + kh * 2;
        int i0 = (k0     < K) ? k0     : 0;
        int i1 = (k0 + 1 < K) ? k0 + 1 : 0;
        float v0 = xrow[i0];
        float v1 = xrow[i1];
        a0[kk] = (k0     < K) ? v0 : 0.0f;
        a1[kk] = (k0 + 1 < K) ? v1 : 0.0f;
        b0[kk] = Wlds[k0][bn];
        b1[kk] = Wlds[k0 + 1][bn];
    }

    v8f acc = {};
#pragma unroll
    for (int kk = 0; kk < KITERS; ++kk) {
        v2f a, b;
        a.x = a0[kk]; a.y = a1[kk];
        b.x = b0[kk]; b.y = b1[kk];
        acc = __builtin_amdgcn_wmma_f32_16x16x4_f32(
            /*neg_a=*/false, a, /*neg_b=*/false, b,
            /*c_mod=*/(short)0, acc, /*reuse_a=*/false, /*reuse_b=*/false);
    }

    if (bn < NC) {
        int mbase = rt * 16 + kh * 8;
#pragma unroll
        for (int r = 0; r < 8; ++r)
            H[(mbase + r) * NC + bn] = acc[r];
    }
}

// ---------------- aggregation ----------------

// O[v,:] = H[v,:] * dis[v]^2   (self-loop term; fully initializes O)
__global__ void self_init_kernel(const float* __restrict__ H,
                                 const float* __restrict__ dis,
                                 float* __restrict__ O,
                                 long long total, int dim) {
    long long idx = (long long)blockIdx.x * blockDim.x + threadIdx.x;
    if (idx >= total) return;
    int v = (int)(idx / dim);
    float s = dis[v];
    O[idx] = H[idx] * s * s;
}

// One thread per (edge, feature-pair). float2 gather (rows are 8B-aligned:
// dim is even), two global_atomic_add_f32 into the L2-resident destination.
__global__ void edge_scatter2_kernel(const float* __restrict__ H,
                                     float* __restrict__ O,
                                     const int* __restrict__ src,
                                     const int* __restrict__ dst,
                                     const float* __restrict__ norm,
                                     long long total2, int dim2) {
    long long idx = (long long)blockIdx.x * blockDim.x + threadIdx.x;
    if (idx >= total2) return;
    int e = (int)(idx / dim2);
    int j = (int)(idx - (long long)e * dim2);
    float w = norm[e];
    const float2* hs = (const float2*)(H + (size_t)src[e] * (size_t)(dim2 * 2));
    float2 v = hs[j];
    float* od = O + (size_t)dst[e] * (size_t)(dim2 * 2) + 2 * j;
    atomicAdd(od,     v.x * w);
    atomicAdd(od + 1, v.y * w);
}

__global__ void bias_relu_kernel(float* __restrict__ O,
                                 const float* __restrict__ b,
                                 long long total, int dim) {
    long long idx = (long long)blockIdx.x * blockDim.x + threadIdx.x;
    if (idx >= total) return;
    int j = (int)(idx % dim);
    O[idx] = fmaxf(O[idx] + b[j], 0.0f);
}

// ---------------- launcher ----------------

static inline size_t align256(size_t x) { return (x + 255) & ~(size_t)255; }

template <int K, int NC>
static void run_layer(const float* in, const float* Wm, const float* bm,
                      const float* dis, const int* src32, const int* dst32,
                      const float* norm, float* h, float* o, hipStream_t stream) {
    const int TB = 256;
    int rtiles = GCN_N_NODES / 16;          // 100000 % 16 == 0
    int ctiles = (NC + 15) / 16;
    int waves  = rtiles * ctiles;
    gemm_wmma_f32_kernel<K, NC><<<(waves + 3) / 4, 128, 0, stream>>>(in, Wm, h, rtiles);

    long long totalN = (long long)GCN_N_NODES * NC;
    self_init_kernel<<<(unsigned)((totalN + TB - 1) / TB), TB, 0, stream>>>(
        h, dis, o, totalN, NC);

    long long totalE2 = (long long)GCN_N_EDGES * (NC / 2);
    edge_scatter2_kernel<<<(unsigned)((totalE2 + TB - 1) / TB), TB, 0, stream>>>(
        h, o, src32, dst32, norm, totalE2, NC / 2);

    bias_relu_kernel<<<(unsigned)((totalN + TB - 1) / TB), TB, 0, stream>>>(
        o, bm, totalN, NC);
}

extern "C" void kernel_launch(void* const* d_in, const int* in_sizes, int n_in,
                              void* d_out, int out_size, void* d_ws, size_t ws_size,
                              hipStream_t stream) {
    const float*     x  = (const float*)d_in[0];
    const long long* ei = (const long long*)d_in[1];  // int64 edge_index [2,E]
    const float*     ew = (const float*)d_in[2];
    const float*     W1 = (const float*)d_in[3];
    const float*     b1 = (const float*)d_in[4];
    const float*     W2 = (const float*)d_in[5];
    const float*     b2 = (const float*)d_in[6];
    const float*     W3 = (const float*)d_in[7];
    const float*     b3 = (const float*)d_in[8];
    float*           out = (float*)d_out;

    char* ws = (char*)d_ws;
    size_t off = 0;
    float* deg   = (float*)(ws + off); off = align256(off + (size_t)GCN_N_NODES * 4);
    float* dis   = (float*)(ws + off); off = align256(off + (size_t)GCN_N_NODES * 4);
    int*   src32 = (int*)  (ws + off); off = align256(off + (size_t)GCN_N_EDGES * 4);
    int*   dst32 = (int*)  (ws + off); off = align256(off + (size_t)GCN_N_EDGES * 4);
    float* norm  = (float*)(ws + off); off = align256(off + (size_t)GCN_N_EDGES * 4);
    float* Hbuf  = (float*)(ws + off); off = align256(off + (size_t)GCN_N_NODES * 108 * 4);
    float* Obuf  = (float*)(ws + off); off = align256(off + (size_t)GCN_N_NODES * 108 * 4);

    const int TB = 256;

    // normalization (once)
    deg_init_kernel<<<(GCN_N_NODES + TB - 1) / TB, TB, 0, stream>>>(deg, GCN_N_NODES);
    deg_scatter_kernel<<<(GCN_N_EDGES + TB - 1) / TB, TB, 0, stream>>>(
        ei + GCN_N_EDGES, ew, deg, GCN_N_EDGES);
    dis_kernel<<<(GCN_N_NODES + TB - 1) / TB, TB, 0, stream>>>(deg, dis, GCN_N_NODES);
    edge_norm_kernel<<<(GCN_N_EDGES + TB - 1) / TB, TB, 0, stream>>>(
        ei, ew, dis, src32, dst32, norm, GCN_N_EDGES);

    // Ping-pong: each gemm reads the previous output before self_init
    // overwrites it (kernels on one stream serialize).
    run_layer< 9,  18>(x,    W1, b1, dis, src32, dst32, norm, Hbuf, Obuf, stream);
    run_layer<18,  54>(Obuf, W2, b2, dis, src32, dst32, norm, Hbuf, Obuf, stream);
    run_layer<54, 108>(Obuf, W3, b3, dis, src32, dst32, norm, Hbuf, out,  stream);
}